// UKF_82025285419124
// MI455X (gfx1250) — compile-verified
//
#include <hip/hip_runtime.h>
#include <cstddef>

typedef float v2f __attribute__((ext_vector_type(2)));
typedef float v8f __attribute__((ext_vector_type(8)));

#if defined(__has_builtin)
#if __has_builtin(__builtin_amdgcn_tensor_load_to_lds) && \
    __has_builtin(__builtin_amdgcn_s_wait_tensorcnt)
#define UKF_TDM 1
#endif
#endif
#ifndef UKF_TDM
#define UKF_TDM 0
#endif

#if UKF_TDM
typedef unsigned int u32x4 __attribute__((ext_vector_type(4)));
typedef int i32x4 __attribute__((ext_vector_type(4)));
typedef int i32x8 __attribute__((ext_vector_type(8)));
#endif

#define NDIM 2048
#define PDIM 512
#define WGT  (1.0f / 4096.0f)   /* w = 1/(2n) */

#define FSQRT(x) __builtin_amdgcn_sqrtf(x)   /* raw v_sqrt_f32 */

// ---------------------------------------------------------------------------
// helpers
// ---------------------------------------------------------------------------
__device__ __forceinline__ float block_reduce_sum(float v, float* red) {
  const int tid = threadIdx.x;
  __syncthreads();
  red[tid] = v;
  __syncthreads();
  for (int s = 128; s > 0; s >>= 1) {
    if (tid < s) red[tid] += red[tid + s];
    __syncthreads();
  }
  return red[0];
}

// ---------------------------------------------------------------------------
// 0) zero workspace
// ---------------------------------------------------------------------------
__global__ __launch_bounds__(256) void zero_ws(float* p, int count) {
  int i = blockIdx.x * 256 + threadIdx.x;
  if (i < count) p[i] = 0.0f;
}

// ---------------------------------------------------------------------------
// 1) xp = A@x + B@u + c1 ; x_est = n*xp      (one row per block)
// ---------------------------------------------------------------------------
__global__ __launch_bounds__(256) void gemv_xp(
    const float* __restrict__ A, const float* __restrict__ B,
    const float* __restrict__ x, const float* __restrict__ u,
    const float* __restrict__ c1, float* __restrict__ xp,
    float* __restrict__ xest) {
  __shared__ float red[256];
  const int row = blockIdx.x;
  float s = 0.0f;
  const float* ar = A + (size_t)row * NDIM;
  for (int k = threadIdx.x; k < NDIM; k += 256) s += ar[k] * x[k];
  const float* br = B + (size_t)row * PDIM;
  for (int k = threadIdx.x; k < PDIM; k += 256) s += br[k] * u[k];
  float tot = block_reduce_sum(s, red);
  if (threadIdx.x == 0) {
    float v = tot + c1[row];
    xp[row] = v;
    xest[row] = (float)NDIM * v;
  }
}

// ---------------------------------------------------------------------------
// 2) du = D@u + c2
// ---------------------------------------------------------------------------
__global__ __launch_bounds__(256) void gemv_du(
    const float* __restrict__ D, const float* __restrict__ u,
    const float* __restrict__ c2, float* __restrict__ du) {
  __shared__ float red[256];
  const int row = blockIdx.x;
  float s = 0.0f;
  const float* dr = D + (size_t)row * PDIM;
  for (int k = threadIdx.x; k < PDIM; k += 256) s += dr[k] * u[k];
  float tot = block_reduce_sum(s, red);
  if (threadIdx.x == 0) du[row] = tot + c2[row];
}

// ---------------------------------------------------------------------------
// 3) y_est = n*(C@x_est + du) ; e = y - y_est
// ---------------------------------------------------------------------------
__global__ __launch_bounds__(256) void gemv_yest(
    const float* __restrict__ C, const float* __restrict__ xest,
    const float* __restrict__ du, const float* __restrict__ y,
    float* __restrict__ yest, float* __restrict__ e) {
  __shared__ float red[256];
  const int row = blockIdx.x;
  float s = 0.0f;
  const float* cr = C + (size_t)row * NDIM;
  for (int k = threadIdx.x; k < NDIM; k += 256) s += cr[k] * xest[k];
  float tot = block_reduce_sum(s, red);
  if (threadIdx.x == 0) {
    float v = (float)NDIM * (tot + du[row]);
    yest[row] = v;
    e[row] = y[row] - v;
  }
}

// ---------------------------------------------------------------------------
// 4) s1 = 2w * sum_{i,j} ((1-n)xp[j] + sqrt(w*P[j,i]))^2    (flat over P)
// ---------------------------------------------------------------------------
__global__ __launch_bounds__(256) void s1_reduce(
    const float* __restrict__ P, const float* __restrict__ xp,
    float* __restrict__ wsS) {
  __shared__ float red[256];
  float acc = 0.0f;
  const size_t total = (size_t)NDIM * NDIM;
  const size_t stride = (size_t)gridDim.x * 256;
  for (size_t idx = (size_t)blockIdx.x * 256 + threadIdx.x; idx < total;
       idx += stride) {
    int row = (int)(idx >> 11);
    float d = (1.0f - (float)NDIM) * xp[row];
    float t = d + FSQRT(WGT * P[idx]);
    acc += t * t;
  }
  float tot = block_reduce_sum(acc, red);
  if (threadIdx.x == 0) atomicAdd(&wsS[0], 2.0f * WGT * tot);
}

// ---------------------------------------------------------------------------
// 5) Fused sigma GEMM:  M = x2pl @ C^T, reduced into s2 / sxy.
//    x2pl[i,k] = x_est[k] + sqrt(w*Q[k,i] + w*s1) built in LDS.
//    Block tile 128(M) x 64(N), BK=16, 8 waves, each wave 32x32 (2x2 WMMA).
//    B tile (C^T panel) is DMA'd by the Tensor Data Mover with LDS padding
//    (16 DW rows + 4 DW pad -> stride-20 conflict-free layout).
// ---------------------------------------------------------------------------
#define TM 128
#define TN 64
#define TK 16

__global__ __launch_bounds__(256) void ukf_sigma_gemm(
    const float* __restrict__ Q, const float* __restrict__ C,
    const float* __restrict__ xest, const float* __restrict__ du,
    const float* __restrict__ yest, float* __restrict__ wsS) {
  __shared__ float lA[TM * 18];     // [ii][kk], padded stride 18
  __shared__ float lB[TN * 20];     // [jj][kk], padded stride 20 (C^T tile)
  __shared__ float lS2[TN * 130];   // [jj][ii], padded stride 130 (epilogue)
  __shared__ float red[512];

  const float s1 = wsS[0];
  const float ws1 = WGT * s1;

  const int tid = threadIdx.x;
  const int lane = tid & 31;
  const int wid = tid >> 5;
  const int wm = wid & 3;                 // 0..3 : 32-row strip
  const int wn = wid >> 2;                // 0..1 : 32-col strip
  const int i0 = blockIdx.x * TM;
  const int j0 = blockIdx.y * TN;

  const int ml = lane & 15;
  const int hh = lane >> 4;               // half select
  const int kh = hh * 2;                  // K sub-offset for A/B fragments

#if UKF_TDM
  // LDS byte offset of lB for the TDM descriptor
  __attribute__((address_space(3))) float* lB3 =
      (__attribute__((address_space(3))) float*)lB;
  const unsigned int lb_addr = (unsigned int)(unsigned long long)lB3;
#endif

  v8f acc[2][2];
  acc[0][0] = v8f{};
  acc[0][1] = v8f{};
  acc[1][0] = v8f{};
  acc[1][1] = v8f{};

  for (int k0 = 0; k0 < NDIM; k0 += TK) {
    __syncthreads();

#if UKF_TDM
    // ---- Tensor Data Mover: DMA the 64x16 f32 C^T panel into lB ----
    if (wid == 0) {
      unsigned long long ga =
          (unsigned long long)(const void*)(C + (size_t)j0 * NDIM + k0);
      u32x4 g0;
      g0[0] = 1u;                                   // count=1 valid user D#
      g0[1] = lb_addr;                              // lds_addr
      g0[2] = (unsigned int)(ga & 0xFFFFFFFFull);   // global_addr lo
      g0[3] = (unsigned int)((ga >> 32) & 0x01FFFFFFull) | (2u << 30);  // type=2
      i32x8 g1;
      g1[0] = (2 << 16)      // data_size = 4B
            | (1 << 20)      // pad_enable
            | (3 << 22)      // pad_interval: 16 DWORDs
            | (3 << 25);     // pad_amount : 4 DWORDs
      g1[1] = (NDIM & 0xFFFF) << 16;   // tensor_dim0 lo16 in bits[63:48]
      g1[2] = (NDIM & 0xFFFF) << 16;   // dim0 hi=0, tensor_dim1 lo16 in [95:80]
      g1[3] = TK << 16;                // tensor_dim1 hi=0, tile_dim0=16
      g1[4] = TN;                      // tile_dim1=64, tile_dim2=0
      g1[5] = NDIM;                    // tensor_dim0_stride = 2048 elements
      g1[6] = 0;
      g1[7] = 0;
      i32x4 gz4 = {0, 0, 0, 0};
      i32x8 gz8 = {0, 0, 0, 0, 0, 0, 0, 0};
      __builtin_amdgcn_tensor_load_to_lds(g0, g1, gz4, gz4, gz8, 0);
    }
#else
    // fallback: cooperative copy of the C^T panel
#pragma unroll
    for (int q = 0; q < 4; ++q) {
      int f = q * 256 + tid;              // 1024 = 64 x 16
      int jj = f >> 4;
      int kk = f & 15;
      lB[jj * 20 + kk] = C[(size_t)(j0 + jj) * NDIM + (k0 + kk)];
    }
#endif

    // A tile: Q rows k (contiguous in i), transform to x2pl, transpose in LDS
#pragma unroll
    for (int q = 0; q < 8; ++q) {
      int f = q * 256 + tid;              // 2048 = 16 x 128
      int kk = f >> 7;
      int ii = f & 127;
      float qv = Q[(size_t)(k0 + kk) * NDIM + (i0 + ii)];
      lA[ii * 18 + kk] = xest[k0 + kk] + FSQRT(WGT * qv + ws1);
    }
    if (k0 + TK < NDIM) {                 // L2 prefetch of next Q panel
      __builtin_prefetch(
          &Q[(size_t)(k0 + TK + (tid >> 4)) * NDIM + i0 + ((tid & 15) << 3)], 0,
          1);
    }

#if UKF_TDM
    if (wid == 0) __builtin_amdgcn_s_wait_tensorcnt(0);
#endif
    __syncthreads();

#pragma unroll
    for (int kb = 0; kb < TK; kb += 4) {
      v2f afr[2], bfr[2];
#pragma unroll
      for (int mt = 0; mt < 2; ++mt) {
        int m = wm * 32 + mt * 16 + ml;
        float2 av = *(const float2*)&lA[m * 18 + kb + kh];
        afr[mt].x = av.x;
        afr[mt].y = av.y;
      }
#pragma unroll
      for (int nt = 0; nt < 2; ++nt) {
        int jj = wn * 32 + nt * 16 + ml;
        float2 bv = *(const float2*)&lB[jj * 20 + kb + kh];
        bfr[nt].x = bv.x;
        bfr[nt].y = bv.y;
      }
#pragma unroll
      for (int mt = 0; mt < 2; ++mt)
#pragma unroll
        for (int nt = 0; nt < 2; ++nt)
          acc[mt][nt] = __builtin_amdgcn_wmma_f32_16x16x4_f32(
              false, afr[mt], false, bfr[nt], (short)0, acc[mt][nt], false,
              false);
    }
  }

  // ---- epilogue: stage S2[i,j] = sqrt(w*Q[j,i] + w*s1) through LDS ----
  __syncthreads();
#pragma unroll
  for (int q = 0; q < 32; ++q) {
    int f = q * 256 + tid;                // 8192 = 64 x 128
    int jj = f >> 7;
    int ii = f & 127;
    float qv = Q[(size_t)(j0 + jj) * NDIM + (i0 + ii)];
    lS2[jj * 130 + ii] = FSQRT(WGT * qv + ws1);
  }
  __syncthreads();

  float s2a = 0.0f, sxya = 0.0f;
#pragma unroll
  for (int nt = 0; nt < 2; ++nt) {
    int jl = wn * 32 + nt * 16 + ml;
    float duv = du[j0 + jl];
    float yev = yest[j0 + jl];
#pragma unroll
    for (int mt = 0; mt < 2; ++mt) {
#pragma unroll
      for (int r = 0; r < 8; ++r) {
        int il = wm * 32 + mt * 16 + r + (hh << 3);
        float dy = (acc[mt][nt][r] + duv) - yev;
        s2a += dy * dy;
        sxya += lS2[jl * 130 + il] * dy;
      }
    }
  }

  red[tid] = s2a;
  red[256 + tid] = sxya;
  __syncthreads();
  for (int s = 128; s > 0; s >>= 1) {
    if (tid < s) {
      red[tid] += red[tid + s];
      red[256 + tid] += red[256 + tid + s];
    }
    __syncthreads();
  }
  if (tid == 0) {
    atomicAdd(&wsS[1], 2.0f * WGT * red[0]);    // s2
    atomicAdd(&wsS[2], 2.0f * WGT * red[256]);  // sxy
  }
}

// ---------------------------------------------------------------------------
// 6) fixed-point sweep for Py z = rhs,  Py = R + s2 (rank-1-shifted GEMV)
// ---------------------------------------------------------------------------
__global__ __launch_bounds__(256) void rich_update(
    const float* __restrict__ R, const float* __restrict__ rhs, int rhs_is_one,
    const float* __restrict__ z, float* __restrict__ znew,
    const float* __restrict__ wsS) {
  __shared__ float red[256];
  const int row = blockIdx.x;
  const float s2 = wsS[1];
  float s = 0.0f;
  const float* rr = R + (size_t)row * NDIM;
  for (int k = threadIdx.x; k < NDIM; k += 256) s += (rr[k] + s2) * z[k];
  float tot = block_reduce_sum(s, red);
  if (threadIdx.x == 0) {
    float rv = rhs_is_one ? 1.0f : rhs[row];
    float omega = 1.0f / ((float)NDIM * s2 + 0.5f * (float)NDIM);
    znew[row] = z[row] + omega * (rv - tot);
  }
}

// ---------------------------------------------------------------------------
// 7) a = sum(z_e), b = sum(z_1); alpha = sxy*a ; psub = s1 - sxy^2*b
// ---------------------------------------------------------------------------
__global__ __launch_bounds__(256) void finalize_scalars(
    const float* __restrict__ ze, const float* __restrict__ z1,
    float* __restrict__ wsS) {
  __shared__ float red[512];
  const int tid = threadIdx.x;
  float a = 0.0f, b = 0.0f;
  for (int k = tid; k < NDIM; k += 256) {
    a += ze[k];
    b += z1[k];
  }
  red[tid] = a;
  red[256 + tid] = b;
  __syncthreads();
  for (int s = 128; s > 0; s >>= 1) {
    if (tid < s) {
      red[tid] += red[tid + s];
      red[256 + tid] += red[256 + tid + s];
    }
    __syncthreads();
  }
  if (tid == 0) {
    float sxy = wsS[2];
    wsS[6] = sxy * red[0];                      // alpha (x shift)
    wsS[7] = wsS[0] - sxy * sxy * red[256];     // s1 - sxy^2*b (P shift)
  }
}

// ---------------------------------------------------------------------------
// 8) outputs: x_new = x_est + alpha ; P_new = Q + psub
// ---------------------------------------------------------------------------
__global__ __launch_bounds__(256) void write_out(
    const float* __restrict__ xest, const float* __restrict__ Q,
    const float* __restrict__ wsS, float* __restrict__ out) {
  size_t idx = (size_t)blockIdx.x * 256 + threadIdx.x;
  const size_t total = (size_t)NDIM + (size_t)NDIM * NDIM;
  if (idx >= total) return;
  if (idx < NDIM)
    out[idx] = xest[idx] + wsS[6];
  else
    out[idx] = Q[idx - NDIM] + wsS[7];
}

// ---------------------------------------------------------------------------
// host launcher
// ---------------------------------------------------------------------------
extern "C" void kernel_launch(void* const* d_in, const int* in_sizes, int n_in,
                              void* d_out, int out_size, void* d_ws,
                              size_t ws_size, hipStream_t stream) {
  const float* x  = (const float*)d_in[0];
  const float* y  = (const float*)d_in[1];
  const float* u  = (const float*)d_in[2];
  const float* P  = (const float*)d_in[3];
  const float* A  = (const float*)d_in[4];
  const float* B  = (const float*)d_in[5];
  const float* C  = (const float*)d_in[6];
  const float* D  = (const float*)d_in[7];
  const float* c1 = (const float*)d_in[8];
  const float* c2 = (const float*)d_in[9];
  const float* Q  = (const float*)d_in[10];
  const float* R  = (const float*)d_in[11];
  float* out = (float*)d_out;
  float* ws  = (float*)d_ws;

  const int ws_floats = 16 + 9 * NDIM;
  if (ws_size < (size_t)ws_floats * sizeof(float)) return;

  float* wsS  = ws;             // [0]=s1 [1]=s2 [2]=sxy [6]=alpha [7]=psub
  float* xp   = ws + 16;
  float* xest = xp + NDIM;
  float* du   = xest + NDIM;
  float* yest = du + NDIM;
  float* e    = yest + NDIM;
  float* zeA  = e + NDIM;
  float* zeB  = zeA + NDIM;
  float* z1A  = zeB + NDIM;
  float* z1B  = z1A + NDIM;

  zero_ws<<<(ws_floats + 255) / 256, 256, 0, stream>>>(ws, ws_floats);
  gemv_xp<<<NDIM, 256, 0, stream>>>(A, B, x, u, c1, xp, xest);
  gemv_du<<<NDIM, 256, 0, stream>>>(D, u, c2, du);
  gemv_yest<<<NDIM, 256, 0, stream>>>(C, xest, du, y, yest, e);
  s1_reduce<<<2048, 256, 0, stream>>>(P, xp, wsS);

  dim3 gg(NDIM / TM, NDIM / TN);  // 16 x 32
  ukf_sigma_gemm<<<gg, 256, 0, stream>>>(Q, C, xest, du, yest, wsS);

  // two approximate solves Py z = rhs (Py = R + s2), 4 sweeps each
  float* za = zeA;
  float* zb = zeB;
  for (int it = 0; it < 4; ++it) {
    rich_update<<<NDIM, 256, 0, stream>>>(R, e, 0, za, zb, wsS);
    float* t = za; za = zb; zb = t;
  }
  float* zeF = za;
  za = z1A;
  zb = z1B;
  for (int it = 0; it < 4; ++it) {
    rich_update<<<NDIM, 256, 0, stream>>>(R, (const float*)0, 1, za, zb, wsS);
    float* t = za; za = zb; zb = t;
  }
  float* z1F = za;

  finalize_scalars<<<1, 256, 0, stream>>>(zeF, z1F, wsS);

  const size_t total = (size_t)NDIM + (size_t)NDIM * NDIM;
  write_out<<<(unsigned)((total + 255) / 256), 256, 0, stream>>>(xest, Q, wsS,
                                                                out);
}